// Solver_53721450938645
// MI455X (gfx1250) — compile-verified
//
#include <hip/hip_runtime.h>
#include <cmath>

// ---------------- problem constants ----------------
#define NXc 240
#define NYc 140
#define NTc 2000
#define DTc 0.005f
#define Hc  0.05f
#define NXY (NXc * NYc)          // 33600
#define NFT (NTc + 2)            // 2002

typedef float v2f __attribute__((ext_vector_type(2)));
typedef float v8f __attribute__((ext_vector_type(8)));

// LDS (address-space 3) helpers for async-to-LDS inline asm
typedef __attribute__((address_space(3))) float lds_f32;

__device__ __forceinline__ unsigned ldsOff(float* p) {
    return (unsigned)(unsigned long long)(lds_f32*)p;
}

// GLOBAL_LOAD_ASYNC_TO_LDS_B32 (GV mode): per-lane global->LDS copy,
// tracked on ASYNCcnt. vdst = LDS byte address, vaddr = 64-bit global addr.
__device__ __forceinline__ void async_ld_lds_b32(unsigned lds, const void* g) {
    asm volatile("global_load_async_to_lds_b32 %0, %1, off"
                 :: "v"(lds), "v"((unsigned long long)g)
                 : "memory");
}

// =====================================================================
// MLP kernel: batched sin-MLP via V_WMMA_F32_16X16X4_F32.
// Each wave processes a tile of 32 points (two 16-row M-tiles) so that
// every B fragment (weights) feeds TWO WMMAs. Hidden layers WIDTHxWIDTH
// are GEMMs D(32xWIDTH) = A(32xK) * Bt(KxWIDTH), K accumulated 4 at a
// time. Activations round-trip through padded LDS (stride WIDTH+1,
// conflict-free) to convert C/D fragment layout -> A layout.
// Weights are staged into LDS with async-to-LDS copies (transposing
// scatter), fenced with s_wait_asynccnt.
// MODE 0: c-field  -> out = (DT/H)^2 * exp(2*mlp)   (coef field)
// MODE 1: source   -> out = mlp                      (f field)
// =====================================================================
template <int WIDTH, int MODE>
__global__ void __launch_bounds__(128)
mlp_wmma_kernel(const float* __restrict__ w0, const float* __restrict__ b0,
                const float* __restrict__ w1, const float* __restrict__ w2,
                const float* __restrict__ w3, const float* __restrict__ w4,
                const float* __restrict__ b1, const float* __restrict__ b2,
                const float* __restrict__ b3, const float* __restrict__ b4,
                const float* __restrict__ w5, const float* __restrict__ b5,
                float* __restrict__ outp, int total)
{
    constexpr int WS  = WIDTH + 1;         // padded stride (bank-conflict free)
    constexpr int NTI = WIDTH / 16;        // N tiles per layer
    extern __shared__ float smem[];

    float* w0s = smem;                      // WIDTH*2
    float* b0s = w0s + WIDTH * 2;           // WIDTH
    float* bts = b0s + WIDTH;               // 4 * WIDTH * WS  (Bt = W^T, padded)
    float* bls = bts + 4 * WIDTH * WS;      // 4 * WIDTH       (hidden biases)
    float* w5s = bls + 4 * WIDTH;           // WIDTH
    float* b5s = w5s + WIDTH;               // 1
    float* acts = b5s + 1;                  // nwaves * 2 * 32 * WS

    const int tid = threadIdx.x;

    // ---- weight staging: async global->LDS with transposing scatter ----
    {
        const float* wl[4] = {w1, w2, w3, w4};
#pragma unroll
        for (int L = 0; L < 4; ++L) {
            const float* wsrc = wl[L];
            float* bdst = bts + L * WIDTH * WS;
            for (int i = tid; i < WIDTH * WIDTH; i += 128) {
                int nn = i / WIDTH, kk = i - nn * WIDTH;    // W[nn][kk]
                async_ld_lds_b32(ldsOff(bdst + kk * WS + nn), wsrc + i);
            }
        }
    }
    // small arrays via regular stores
    for (int i = tid; i < WIDTH * 2; i += 128) w0s[i] = w0[i];
    for (int i = tid; i < WIDTH; i += 128) {
        b0s[i] = b0[i];
        w5s[i] = w5[i];
        bls[0 * WIDTH + i] = b1[i];
        bls[1 * WIDTH + i] = b2[i];
        bls[2 * WIDTH + i] = b3[i];
        bls[3 * WIDTH + i] = b4[i];
    }
    if (tid == 0) b5s[0] = b5[0];
    asm volatile("s_wait_asynccnt 0" ::: "memory");
    __syncthreads();

    const int lane = tid & 31;
    const int wave = tid >> 5;
    const int m    = lane & 15;      // row within a 16-row M-tile
    const int half = lane >> 4;      // 0: K+0/1 cols, 1: K+2/3 cols
    float* actA = acts + wave * (2 * 32 * WS);
    float* actB = actA + 32 * WS;

    const int gwave  = blockIdx.x * 4 + wave;
    const int nwaves = gridDim.x * 4;
    const int ntiles = (total + 31) >> 5;   // 32 points per tile

    for (int tile = gwave; tile < ntiles; tile += nwaves) {
        const int base = tile * 32;

        // ---- layer 0 : 2 -> WIDTH, sin activation (two rows per lane) ----
#pragma unroll
        for (int sub = 0; sub < 2; ++sub) {
            int row = m + 16 * sub;
            int p = base + row;
            if (p >= total) p = total - 1;   // clamp (stores guarded later)
            float in0, in1;
            if (MODE == 0) {
                int iy = p / NXc; int ix = p - iy * NXc;
                in0 = Hc * (float)ix;
                in1 = Hc * (float)(NYc - 1 - iy);
            } else {
                int it = p / NXc; int ix = p - it * NXc;
                in0 = Hc * (float)ix;
                in1 = DTc * (float)(it - 2);
            }
            for (int n = half * (WIDTH / 2); n < (half + 1) * (WIDTH / 2); ++n)
                actA[row * WS + n] =
                    __sinf(in0 * w0s[2 * n] + in1 * w0s[2 * n + 1] + b0s[n]);
        }

        float* src = actA;
        float* dst = actB;

        // ---- hidden layers 1..4 : WIDTH -> WIDTH via WMMA f32 ----
        for (int L = 0; L < 4; ++L) {
            asm volatile("s_wait_dscnt 0" ::: "memory");
            const float* BT = bts + L * WIDTH * WS;
            const float* bb = bls + L * WIDTH;

            v8f acc0[NTI], acc1[NTI];
#pragma unroll
            for (int nt = 0; nt < NTI; ++nt) {
                acc0[nt] = (v8f){0,0,0,0,0,0,0,0};
                acc1[nt] = (v8f){0,0,0,0,0,0,0,0};
            }

            for (int kb = 0; kb < WIDTH; kb += 4) {
                // A fragments for both M-tiles
                // lanes0-15 -> K=kb,kb+1 ; lanes16-31 -> kb+2,kb+3
                v2f a0, a1;
                a0.x = src[m * WS + kb + half * 2 + 0];
                a0.y = src[m * WS + kb + half * 2 + 1];
                a1.x = src[(m + 16) * WS + kb + half * 2 + 0];
                a1.y = src[(m + 16) * WS + kb + half * 2 + 1];
#pragma unroll
                for (int nt = 0; nt < NTI; ++nt) {
                    int n  = nt * 16 + m;
                    int kr = kb + half * 2;
                    v2f b;
                    b.x = BT[(kr + 0) * WS + n];
                    b.y = BT[(kr + 1) * WS + n];
                    acc0[nt] = __builtin_amdgcn_wmma_f32_16x16x4_f32(
                        false, a0, false, b, (short)0, acc0[nt], false, false);
                    acc1[nt] = __builtin_amdgcn_wmma_f32_16x16x4_f32(
                        false, a1, false, b, (short)0, acc1[nt], false, false);
                }
            }
            // bias + sin, write back in C/D layout: VGPR r -> M = r + 8*half
#pragma unroll
            for (int nt = 0; nt < NTI; ++nt) {
                int n = nt * 16 + m;
                float bias = bb[n];
#pragma unroll
                for (int r = 0; r < 8; ++r) {
                    dst[(r + 8 * half) * WS + n]      = __sinf(acc0[nt][r] + bias);
                    dst[(16 + r + 8 * half) * WS + n] = __sinf(acc1[nt][r] + bias);
                }
            }
            float* t = src; src = dst; dst = t;
        }

        asm volatile("s_wait_dscnt 0" ::: "memory");
        // ---- final layer WIDTH -> 1 : all 32 lanes, one row each ----
        {
            float sum = b5s[0];
            for (int k = 0; k < WIDTH; ++k)
                sum += src[lane * WS + k] * w5s[k];
            int pp = base + lane;
            if (pp < total) {
                float o;
                if (MODE == 0)
                    o = (DTc * DTc / (Hc * Hc)) * __expf(2.0f * sum); // (DT*exp(mlp)/H)^2
                else
                    o = sum;
                outp[pp] = o;
            }
        }
    }
}

// =====================================================================
// Time-stepping kernel: whole 140x240 grid lives in LDS (2 ping-pong
// buffers = 268.8 KB < 320 KB/WGP). One 1024-thread workgroup runs all
// 2000 steps; coef held in registers (33 cells/thread). Only HBM
// traffic is the mandatory 268.8 MB output stream. Next step's source
// row is prefetched (global_prefetch_b8) while the current step runs.
// =====================================================================
__global__ void __launch_bounds__(1024)
wave_lds_kernel(const float* __restrict__ coef,
                const float* __restrict__ f,     // (NT+2) x NX
                float* __restrict__ out)          // NT x NY x NX
{
    extern __shared__ float smem[];
    float* buf0 = smem;          // NXY floats
    float* buf1 = smem + NXY;    // NXY floats

    const int tid = threadIdx.x;
    constexpr int CPT = (NXY + 1023) / 1024;   // 33 cells per thread

    float rc[CPT];
#pragma unroll
    for (int j = 0; j < CPT; ++j) {
        int c = tid + j * 1024;
        rc[j] = (c < NXY) ? coef[c] : 0.0f;
    }

    // u0 = zeros with last two rows = f[0]; u1 likewise with f[1]
#pragma unroll
    for (int j = 0; j < CPT; ++j) {
        int c = tid + j * 1024;
        if (c < NXY) {
            float v0 = 0.0f, v1 = 0.0f;
            if (c >= (NYc - 2) * NXc) {
                int ix = c % NXc;
                v0 = f[0 * NXc + ix];
                v1 = f[1 * NXc + ix];
            }
            buf0[c] = v0;   // u_prev
            buf1[c] = v1;   // u_cur
        }
    }
    __syncthreads();

    float* up = buf0;
    float* uc = buf1;

    for (int s = 0; s < NTc; ++s) {
        // prefetch next step's source row into cache while we compute
        if (s + 3 < NFT && tid < 8)
            __builtin_prefetch((const char*)(f + (size_t)(s + 3) * NXc) + tid * 128, 0, 3);

        const float* fs = f + (size_t)(s + 2) * NXc;
        float* orow = out + (size_t)s * NXY;
#pragma unroll 1
        for (int j = 0; j < CPT; ++j) {
            int c = tid + j * 1024;
            if (c < NXY) {
                int iy = c / NXc;
                int ix = c - iy * NXc;
                float u  = uc[c];
                float ul = uc[(ix == 0)        ? c : c - 1];
                float ur = uc[(ix == NXc - 1)  ? c : c + 1];
                float ud = uc[(iy == 0)        ? c : c - NXc];
                float uu = uc[(iy == NYc - 1)  ? c : c + NXc];
                float lap = ul + ur + ud + uu - 4.0f * u;
                float un = -up[c] + 2.0f * u + rc[j] * lap;
                if (iy >= NYc - 2) un = fs[ix];     // forced source rows
                up[c] = un;                         // becomes next u_cur
                orow[c] = un;
            }
        }
        __syncthreads();
        float* t = up; up = uc; uc = t;
    }
}

// =====================================================================
// host launcher
// =====================================================================
extern "C" void kernel_launch(void* const* d_in, const int* in_sizes, int n_in,
                              void* d_out, int out_size, void* d_ws, size_t ws_size,
                              hipStream_t stream)
{
    (void)in_sizes; (void)n_in; (void)out_size; (void)ws_size;

    // inputs: cw0,cb0,cw1,cb1,...,cw5,cb5, sw0,sb0,...,sw5,sb5
    const float* cw[6]; const float* cb[6];
    const float* sw[6]; const float* sb[6];
    for (int i = 0; i < 6; ++i) {
        cw[i] = (const float*)d_in[2 * i + 0];
        cb[i] = (const float*)d_in[2 * i + 1];
        sw[i] = (const float*)d_in[12 + 2 * i + 0];
        sb[i] = (const float*)d_in[12 + 2 * i + 1];
    }

    float* ws   = (float*)d_ws;
    float* coef = ws;            // NXY floats
    float* fall = ws + NXY;      // NFT*NX floats

    // dynamic LDS sizes
    auto smemBytes = [](int W, int nwaves) {
        int WS = W + 1;
        int fl = W * 2 + W + 4 * W * WS + 4 * W + W + 1 + nwaves * 2 * 32 * WS;
        return (size_t)fl * sizeof(float);
    };
    size_t sm64 = smemBytes(64, 4);                // ~135 KB
    size_t sm32 = smemBytes(32, 4);                // ~52 KB
    size_t smWv = (size_t)2 * NXY * sizeof(float); // 268.8 KB

    hipFuncSetAttribute((const void*)&mlp_wmma_kernel<64, 0>,
                        hipFuncAttributeMaxDynamicSharedMemorySize, (int)sm64);
    hipFuncSetAttribute((const void*)&mlp_wmma_kernel<32, 1>,
                        hipFuncAttributeMaxDynamicSharedMemorySize, (int)sm32);
    hipFuncSetAttribute((const void*)&wave_lds_kernel,
                        hipFuncAttributeMaxDynamicSharedMemorySize, (int)smWv);

    // 1) coef field: 33600 points, 1050 tiles of 32
    mlp_wmma_kernel<64, 0><<<256, 128, sm64, stream>>>(
        cw[0], cb[0], cw[1], cw[2], cw[3], cw[4],
        cb[1], cb[2], cb[3], cb[4], cw[5], cb[5],
        coef, NXY);

    // 2) source field: 2002*240 points, 15015 tiles of 32
    mlp_wmma_kernel<32, 1><<<512, 128, sm32, stream>>>(
        sw[0], sb[0], sw[1], sw[2], sw[3], sw[4],
        sb[1], sb[2], sb[3], sb[4], sw[5], sb[5],
        fall, NFT * NXc);

    // 3) 2000-step wave propagation, whole grid resident in one WGP's LDS
    wave_lds_kernel<<<1, 1024, smWv, stream>>>(coef, fall, (float*)d_out);
}